// TransformerLayer_4896262717846
// MI455X (gfx1250) — compile-verified
//
#include <hip/hip_runtime.h>
#include <hip/hip_bf16.h>

typedef float v2f __attribute__((ext_vector_type(2)));
typedef float v8f __attribute__((ext_vector_type(8)));
typedef unsigned int u32x4 __attribute__((ext_vector_type(4)));
typedef int i32x8 __attribute__((ext_vector_type(8)));
typedef int i32x4 __attribute__((ext_vector_type(4)));

#define B_   16
#define T_   1024
#define D_   512
#define H_   8
#define DH_  64
#define FF_  2048
#define NQKV 1536
#define KC   32   // K-chunk staged per TDM transfer

static __device__ __forceinline__ v8f wmma4(v2f a, v2f b, v8f c) {
  // V_WMMA_F32_16X16X4_F32: D = A(16x4 f32) * B(4x16 f32) + C(16x16 f32)
  return __builtin_amdgcn_wmma_f32_16x16x4_f32(false, a, false, b, (short)0, c,
                                               false, false);
}

// ---------------------------------------------------------------------------
// TDM 2D tile load: rows x cols fp32 tile, row stride rowStride (elements),
// global -> LDS at byte offset ldsOff. Descriptor per CDNA5 ISA sec 8.3/8.4.
// Tensor dims set huge so no OOB clipping (tiles always in-bounds here).
// 6-arg builtin variant (group2, group3, group4(=i32x8), cpol).
// ---------------------------------------------------------------------------
static __device__ __forceinline__ void tdm_load_2d(unsigned ldsOff,
                                                   const float* gaddr,
                                                   int rows, int cols,
                                                   long rowStride) {
  const unsigned long long ga = (unsigned long long)(uintptr_t)gaddr;
  u32x4 g0;
  g0[0] = 1u;                                   // count=1, user mode
  g0[1] = ldsOff;                               // lds_addr (bytes)
  g0[2] = (unsigned)ga;                         // global_addr[31:0]
  g0[3] = (unsigned)((ga >> 32) & 0x01FFFFFFu)  // global_addr[56:32]
          | (2u << 30);                         // type = 2 ("image")
  const unsigned td0 = 0x40000000u;             // tensor_dim0 (huge)
  const unsigned td1 = 0x00100000u;             // tensor_dim1 (huge)
  const unsigned long long st0 = (unsigned long long)rowStride;
  i32x8 g1;
  g1[0] = (int)(2u << 16);                                          // data_size=4B
  g1[1] = (int)((td0 & 0xFFFFu) << 16);                             // dim0 lo
  g1[2] = (int)(((td0 >> 16) & 0xFFFFu) | ((td1 & 0xFFFFu) << 16)); // dim0 hi|dim1 lo
  g1[3] = (int)(((td1 >> 16) & 0xFFFFu) | ((unsigned)cols << 16));  // dim1 hi|tile_dim0
  g1[4] = (int)((unsigned)rows & 0xFFFFu);                          // tile_dim1, tile_dim2=0
  g1[5] = (int)(unsigned)(st0 & 0xFFFFFFFFu);                       // dim0_stride lo
  g1[6] = (int)(unsigned)((st0 >> 32) & 0xFFFFu);                   // dim0_stride hi
  g1[7] = 0;
  const i32x4 z4 = {0, 0, 0, 0};
  const i32x8 z8 = {0, 0, 0, 0, 0, 0, 0, 0};
  __builtin_amdgcn_tensor_load_to_lds(g0, g1, z4, z4, z8, 0);
}

static __device__ __forceinline__ unsigned lds_off(const void* p) {
  return (unsigned)(uintptr_t)p;  // low 32 bits of generic ptr = LDS offset
}

// ---------------------------------------------------------------------------
// GEMM: C[m][n] = A[m][k] * W[n][k] (+bias) (+residual). M-tile 32, N-tile 64,
// one wave32 per block. W tiles staged to LDS by TDM, double buffered.
// ---------------------------------------------------------------------------
__global__ __launch_bounds__(32) void gemm_wmma(
    const float* __restrict__ A, const float* __restrict__ W,
    const float* __restrict__ bias, const float* __restrict__ residual,
    float* __restrict__ C, int Kd, int N) {
  const int lane = threadIdx.x;
  const int m0 = blockIdx.x * 32;
  const int n0 = blockIdx.y * 64;
  const int hl = lane & 15;
  const int koff = (lane >> 4) * 2;   // lanes 16-31 hold K+2,K+3
  v8f acc[2][4] = {};

  __shared__ float sW[2][64 * KC];    // 16 KB double buffer

  const float* a0p = A + (size_t)(m0 + hl) * Kd + koff;
  const float* a1p = a0p + (size_t)16 * Kd;
  const float* wtile = W + (size_t)n0 * Kd;
  const int nch = Kd / KC;

  tdm_load_2d(lds_off(&sW[0][0]), wtile, 64, KC, Kd);

  for (int ci = 0; ci < nch; ++ci) {
    const int k = ci * KC;
    const int nb = ci & 1;
    // batch A fragment loads for this chunk (pipeline with TDM wait)
    v2f a0[KC / 4], a1[KC / 4];
#pragma unroll
    for (int s = 0; s < KC / 4; ++s) {
      a0[s] = *(const v2f*)(a0p + k + 4 * s);
      a1[s] = *(const v2f*)(a1p + k + 4 * s);
    }
    if (ci + 1 < nch) {
      tdm_load_2d(lds_off(&sW[nb ^ 1][0]), wtile + k + KC, 64, KC, Kd);
      __builtin_amdgcn_s_wait_tensorcnt(1);
    } else {
      __builtin_amdgcn_s_wait_tensorcnt(0);
    }
    const float* wl = &sW[nb][0];
#pragma unroll
    for (int s = 0; s < KC / 4; ++s) {
#pragma unroll
      for (int j = 0; j < 4; ++j) {
        v2f wb = *(const v2f*)(wl + (j * 16 + hl) * KC + 4 * s + koff);
        acc[0][j] = wmma4(a0[s], wb, acc[0][j]);
        acc[1][j] = wmma4(a1[s], wb, acc[1][j]);
      }
    }
  }

  const int rbase = (lane >> 4) * 8;  // C/D layout: lanes 16-31 hold M=r+8
#pragma unroll
  for (int s = 0; s < 2; ++s) {
#pragma unroll
    for (int j = 0; j < 4; ++j) {
      const int col = n0 + j * 16 + hl;
      const float bv = bias ? bias[col] : 0.f;
#pragma unroll
      for (int r = 0; r < 8; ++r) {
        const int row = m0 + s * 16 + rbase + r;
        float v = acc[s][j][r] + bv;
        if (residual) v += residual[(size_t)row * N + col];
        C[(size_t)row * N + col] = v;
      }
    }
  }
}

// ---------------------------------------------------------------------------
// Conv1d (K=3, pad 1) as 3-tap WMMA GEMM with repacked weights
// Wr[tap][cout][cin] (contiguous rows -> TDM-stageable).
// ---------------------------------------------------------------------------
__global__ __launch_bounds__(32) void conv_wmma(
    const float* __restrict__ X, const float* __restrict__ Wr,
    const float* __restrict__ bias, const float* __restrict__ residual,
    float* __restrict__ C, int Cin, int Cout, int doRelu) {
  const int lane = threadIdx.x;
  const int m0 = blockIdx.x * 32;  // rows = b*T + t; tiles never cross batch
  const int n0 = blockIdx.y * 64;
  const int hl = lane & 15;
  const int koff = (lane >> 4) * 2;
  v8f acc[2][4] = {};
  const v2f zero2 = {0.f, 0.f};

  __shared__ float sW[2][64 * KC];

  const int r0 = m0 + hl;
  const int r1 = r0 + 16;
  const int t0 = r0 & (T_ - 1);
  const int t1 = r1 & (T_ - 1);
  const int nch = Cin / KC;
  const int total = 3 * nch;

  // chunk cc -> (tap = cc/nch, k = (cc%nch)*KC)
  tdm_load_2d(lds_off(&sW[0][0]), Wr + (size_t)n0 * Cin, 64, KC, Cin);

  for (int cc = 0; cc < total; ++cc) {
    const int tap = cc / nch;
    const int k = (cc - tap * nch) * KC;
    const int nb = cc & 1;
    const int sh = tap - 1;
    const bool v0 = ((unsigned)(t0 + sh) < (unsigned)T_);
    const bool v1 = ((unsigned)(t1 + sh) < (unsigned)T_);
    const float* x0 = X + (size_t)(r0 + sh) * Cin + k + koff;
    const float* x1 = X + (size_t)(r1 + sh) * Cin + k + koff;
    v2f a0[KC / 4], a1[KC / 4];
#pragma unroll
    for (int s = 0; s < KC / 4; ++s) {
      a0[s] = v0 ? *(const v2f*)(x0 + 4 * s) : zero2;
      a1[s] = v1 ? *(const v2f*)(x1 + 4 * s) : zero2;
    }
    if (cc + 1 < total) {
      const int tap2 = (cc + 1) / nch;
      const int k2 = ((cc + 1) - tap2 * nch) * KC;
      const float* wt2 = Wr + ((size_t)tap2 * Cout + n0) * Cin + k2;
      tdm_load_2d(lds_off(&sW[nb ^ 1][0]), wt2, 64, KC, Cin);
      __builtin_amdgcn_s_wait_tensorcnt(1);
    } else {
      __builtin_amdgcn_s_wait_tensorcnt(0);
    }
    const float* wl = &sW[nb][0];
#pragma unroll
    for (int s = 0; s < KC / 4; ++s) {
#pragma unroll
      for (int j = 0; j < 4; ++j) {
        v2f wb = *(const v2f*)(wl + (j * 16 + hl) * KC + 4 * s + koff);
        acc[0][j] = wmma4(a0[s], wb, acc[0][j]);
        acc[1][j] = wmma4(a1[s], wb, acc[1][j]);
      }
    }
  }

  const int rbase = (lane >> 4) * 8;
#pragma unroll
  for (int s = 0; s < 2; ++s) {
#pragma unroll
    for (int j = 0; j < 4; ++j) {
      const int col = n0 + j * 16 + hl;
      const float bv = bias ? bias[col] : 0.f;
#pragma unroll
      for (int r = 0; r < 8; ++r) {
        const int row = m0 + s * 16 + rbase + r;
        float v = acc[s][j][r] + bv;
        if (doRelu) v = fmaxf(v, 0.f);
        if (residual) v += residual[(size_t)row * Cout + col];
        C[(size_t)row * Cout + col] = v;
      }
    }
  }
}

// ---------------------------------------------------------------------------
// Weight repack: src[f][c][tap] -> dst[tap][f][c]
// ---------------------------------------------------------------------------
__global__ __launch_bounds__(256) void repack_w(const float* __restrict__ src,
                                                float* __restrict__ dst,
                                                int Cout, int Cin) {
  const size_t i = (size_t)blockIdx.x * 256 + threadIdx.x;
  const size_t total = (size_t)Cout * Cin * 3;
  if (i >= total) return;
  const int tap = (int)(i % 3);
  const size_t fc = i / 3;
  const int c = (int)(fc % Cin);
  const int f = (int)(fc / Cin);
  dst[((size_t)tap * Cout + f) * Cin + c] = src[i];
}

// ---------------------------------------------------------------------------
// Flash attention: one wave per (b*H+h, 16-query tile). Faithful to the
// reference quirks: mask batch = (b*H+h)%B, output goes to
// (batch = i%B, head slot = i/B).
// ---------------------------------------------------------------------------
__global__ __launch_bounds__(32) void attn_wmma(
    const float* __restrict__ qkv, const unsigned char* __restrict__ mask,
    float* __restrict__ vec) {
  const int lane = threadIdx.x;
  const int bh = blockIdx.x;          // b*H + h
  const int qt = blockIdx.y;
  const int b = bh >> 3;              // H = 8
  const int h = bh & 7;
  const int mb = bh & (B_ - 1);       // mask batch quirk
  const int hl = lane & 15;
  const int koff = (lane >> 4) * 2;
  const int t0 = qt * 16;
  const float NEG_INF = -__builtin_inff();

  __shared__ float sP[16][17];

  const float* qrow = qkv + (size_t)(b * T_ + t0 + hl) * NQKV + h * DH_ + koff;
  v2f qa[16];
#pragma unroll
  for (int s = 0; s < 16; ++s) qa[s] = *(const v2f*)(qrow + 4 * s);

  v8f o[4] = {};
  float mrow[8], lrow[8];
#pragma unroll
  for (int r = 0; r < 8; ++r) { mrow[r] = NEG_INF; lrow[r] = 0.f; }

  const float scale = 0.125f;  // DH^-0.5

  for (int kt = 0; kt < T_ / 16; ++kt) {
    const float* krow =
        qkv + (size_t)(b * T_ + kt * 16 + hl) * NQKV + 512 + h * DH_ + koff;
    v8f S = {};
#pragma unroll
    for (int s = 0; s < 16; ++s) {
      v2f kb = *(const v2f*)(krow + 4 * s);
      S = wmma4(qa[s], kb, S);
    }

    const bool valid = mask[(size_t)mb * T_ + kt * 16 + hl] != 0;
    float p[8], alpha[8];
#pragma unroll
    for (int r = 0; r < 8; ++r) {
      float sv = valid ? S[r] * scale : NEG_INF;
      float rm = sv;
#pragma unroll
      for (int m = 1; m < 16; m <<= 1) rm = fmaxf(rm, __shfl_xor(rm, m, 32));
      const float nm = fmaxf(mrow[r], rm);
      alpha[r] = __expf(mrow[r] - nm);
      const float pv = __expf(sv - nm);
      float rs = pv;
#pragma unroll
      for (int m = 1; m < 16; m <<= 1) rs += __shfl_xor(rs, m, 32);
      lrow[r] = lrow[r] * alpha[r] + rs;
      mrow[r] = nm;
      p[r] = pv;
    }
#pragma unroll
    for (int j = 0; j < 4; ++j)
#pragma unroll
      for (int r = 0; r < 8; ++r) o[j][r] *= alpha[r];

    const int rr0 = (lane >> 4) * 8;
#pragma unroll
    for (int r = 0; r < 8; ++r) sP[rr0 + r][hl] = p[r];
    __syncthreads();

    const float* vbase = qkv + (size_t)(b * T_ + kt * 16) * NQKV + 1024 + h * DH_;
#pragma unroll
    for (int s = 0; s < 4; ++s) {
      v2f pa;
      pa.x = sP[hl][4 * s + koff];
      pa.y = sP[hl][4 * s + koff + 1];
#pragma unroll
      for (int j = 0; j < 4; ++j) {
        const int dim = j * 16 + hl;
        v2f vb;
        vb.x = vbase[(size_t)(4 * s + koff) * NQKV + dim];
        vb.y = vbase[(size_t)(4 * s + koff + 1) * NQKV + dim];
        o[j] = wmma4(pa, vb, o[j]);
      }
    }
    __syncthreads();
  }

  const int ob = bh & (B_ - 1);  // output batch = i % B
  const int hs = bh >> 4;        // head slot    = i / B
  float inv[8];
#pragma unroll
  for (int r = 0; r < 8; ++r) inv[r] = 1.f / lrow[r];
  const int rbase = (lane >> 4) * 8;
#pragma unroll
  for (int j = 0; j < 4; ++j) {
    const int col = hs * DH_ + j * 16 + hl;
#pragma unroll
    for (int r = 0; r < 8; ++r) {
      const int row = t0 + rbase + r;
      vec[(size_t)(ob * T_ + row) * (H_ * DH_) + col] = o[j][r] * inv[r];
    }
  }
}

// ---------------------------------------------------------------------------
// LayerNorm(+mask) over rows of D=512. One block (8 waves) per row.
// ---------------------------------------------------------------------------
__global__ __launch_bounds__(256) void ln_kernel(
    const float* __restrict__ X, const float* __restrict__ g,
    const float* __restrict__ bt, const unsigned char* __restrict__ mask,
    float* __restrict__ out) {
  const int row = blockIdx.x;
  const int tid = threadIdx.x;
  const float x0 = X[(size_t)row * D_ + tid];
  const float x1 = X[(size_t)row * D_ + tid + 256];
  float s = x0 + x1;
  float sq = x0 * x0 + x1 * x1;
#pragma unroll
  for (int m = 1; m < 32; m <<= 1) {
    s += __shfl_xor(s, m, 32);
    sq += __shfl_xor(sq, m, 32);
  }
  __shared__ float ssum[8], ssq[8];
  const int w = tid >> 5;
  if ((tid & 31) == 0) { ssum[w] = s; ssq[w] = sq; }
  __syncthreads();
  s = 0.f; sq = 0.f;
#pragma unroll
  for (int i = 0; i < 8; ++i) { s += ssum[i]; sq += ssq[i]; }
  const float mu = s * (1.f / D_);
  const float var = sq * (1.f / D_) - mu * mu;
  const float rstd = rsqrtf(var + 1e-5f);
  const float mv = mask[row] ? 1.f : 0.f;
  out[(size_t)row * D_ + tid] = ((x0 - mu) * rstd * g[tid] + bt[tid]) * mv;
  out[(size_t)row * D_ + tid + 256] =
      ((x1 - mu) * rstd * g[tid + 256] + bt[tid + 256]) * mv;
}

// ---------------------------------------------------------------------------
extern "C" void kernel_launch(void* const* d_in, const int* in_sizes, int n_in,
                              void* d_out, int out_size, void* d_ws,
                              size_t ws_size, hipStream_t stream) {
  (void)in_sizes; (void)n_in; (void)out_size; (void)ws_size;
  const float* dec_inp = (const float*)d_in[0];
  const unsigned char* maskp = (const unsigned char*)d_in[1];
  const float* qkv_w = (const float*)d_in[2];
  const float* qkv_b = (const float*)d_in[3];
  const float* o_w   = (const float*)d_in[4];
  const float* ln1_g = (const float*)d_in[5];
  const float* ln1_b = (const float*)d_in[6];
  const float* c1_w  = (const float*)d_in[7];
  const float* c1_b  = (const float*)d_in[8];
  const float* c2_w  = (const float*)d_in[9];
  const float* c2_b  = (const float*)d_in[10];
  const float* ln2_g = (const float*)d_in[11];
  const float* ln2_b = (const float*)d_in[12];
  float* outp = (float*)d_out;

  // Workspace plan (bytes):
  //   region A [0,       134.2M): qkv -> res1 -> conv hidden (sequential reuse)
  //   region B [134.2M,  167.8M): vec
  //   region C [167.8M,  201.3M): out1
  //   region D [201.3M,  213.9M): w1 repacked [3][FF][D]
  //   region E [213.9M,  226.5M): w2 repacked [3][D][FF]
  float* ws   = (float*)d_ws;
  float* qkvb = ws;
  float* vecb = ws + (size_t)134217728 / 4;
  float* res1 = ws;
  float* out1 = ws + (size_t)167772160 / 4;
  float* hbuf = ws;
  float* w1r  = ws + (size_t)201326592 / 4;
  float* w2r  = ws + (size_t)213909504 / 4;

  const int M = B_ * T_;  // 16384

  // 0) repack conv weights to [tap][cout][cin]
  {
    const size_t n1 = (size_t)FF_ * D_ * 3;
    repack_w<<<(unsigned)((n1 + 255) / 256), 256, 0, stream>>>(c1_w, w1r, FF_, D_);
    const size_t n2 = (size_t)D_ * FF_ * 3;
    repack_w<<<(unsigned)((n2 + 255) / 256), 256, 0, stream>>>(c2_w, w2r, D_, FF_);
  }
  // 1) QKV projection
  gemm_wmma<<<dim3(M / 32, NQKV / 64), 32, 0, stream>>>(
      dec_inp, qkv_w, qkv_b, nullptr, qkvb, D_, NQKV);
  // 2) attention
  attn_wmma<<<dim3(B_ * H_, T_ / 16), 32, 0, stream>>>(qkvb, maskp, vecb);
  // 3) output projection + residual
  gemm_wmma<<<dim3(M / 32, D_ / 64), 32, 0, stream>>>(
      vecb, o_w, nullptr, dec_inp, res1, H_ * DH_, D_);
  // 4) LN1 + mask
  ln_kernel<<<M, 256, 0, stream>>>(res1, ln1_g, ln1_b, maskp, out1);
  // 5) conv1 + relu
  conv_wmma<<<dim3(M / 32, FF_ / 64), 32, 0, stream>>>(
      out1, w1r, c1_b, nullptr, hbuf, D_, FF_, 1);
  // 6) conv2 + residual (into d_out)
  conv_wmma<<<dim3(M / 32, D_ / 64), 32, 0, stream>>>(
      hbuf, w2r, c2_b, out1, outp, FF_, D_, 0);
  // 7) LN2 + mask, in place on d_out
  ln_kernel<<<M, 256, 0, stream>>>(outp, ln2_g, ln2_b, maskp, outp);
}